// ChamferCUDA_60241211293823
// MI455X (gfx1250) — compile-verified
//
#include <hip/hip_runtime.h>
#include <math.h>

typedef __attribute__((ext_vector_type(2))) float v2f;
typedef __attribute__((ext_vector_type(8))) float v8f;

#define EPSF 1e-6f
#define COLSPLIT 4   // column-range splits per 16-row strip (occupancy knob)

// Order-preserving map: float bits -> u32 key, ascending numeric order.
__device__ __forceinline__ unsigned int orderKey(float f) {
  unsigned int b = __float_as_uint(f);
  return (b & 0x80000000u) ? ~b : (b | 0x80000000u);
}
__device__ __forceinline__ float keyToFloat(unsigned int k) {
  unsigned int b = (k & 0x80000000u) ? (k & 0x7FFFFFFFu) : ~k;
  return __uint_as_float(b);
}

// ---------------------------------------------------------------------------
// Each wave owns a 16-row strip of X and a contiguous 1/COLSPLIT chunk of Y.
// One V_WMMA_F32_16X16X4_F32 per 16x16 tile computes the PARTIAL distance
//   D[m][n] = |y_n|^2 - 2 x_m.y_n            (row-constant |x_m|^2 omitted)
//   A (16x4) = [x, y, z, 1]        (lanes 0-15: K0,K1 ; lanes 16-31: K2,K3)
//   B (4x16) = [-2yx, -2yy, -2yz, |y|^2]
//   C        = 0  (inline SRC2=0)
// Steady-state loop: 4 tiles/iteration, two ping-pong buffers, loads land in
// the registers consumed next iteration, NO branches in the body; the final
// 4 buffered tiles are processed in a peeled epilogue.
// Partial (min,argmin) merged across chunks with packed u64 atomicMin.
// ---------------------------------------------------------------------------
__global__ __launch_bounds__(256) void chamfer_nn_wmma(
    const float* __restrict__ X,            // [B,N,3]
    const float* __restrict__ Y,            // [B,M,3]
    unsigned long long* __restrict__ best,  // [B,N] packed (key<<32 | idx)
    int N, int M, int totalWaves)
{
  const int lane = threadIdx.x & 31;        // wave32
  const int wid  = blockIdx.x * (blockDim.x >> 5) + (threadIdx.x >> 5);
  if (wid >= totalWaves) return;

  const int chunk = wid % COLSPLIT;
  const int strip = wid / COLSPLIT;
  const int stripsPerBatch = N >> 4;
  const int b        = strip / stripsPerBatch;
  const int rowBase  = (strip % stripsPerBatch) << 4;
  const int colsPerChunk = M / COLSPLIT;
  const int colStart = chunk * colsPerChunk;
  const int numTiles = colsPerChunk >> 4;   // 128 for the reference shape

  const float* xb = X + (size_t)b * N * 3;
  const float* yb = Y + (size_t)b * M * 3;

  // ---- A fragment (constant for the whole strip) ----
  const int  r  = lane & 15;
  const bool hi = lane >= 16;
  const float px = xb[(rowBase + r) * 3 + 0];
  const float py = xb[(rowBase + r) * 3 + 1];
  const float pz = xb[(rowBase + r) * 3 + 2];
  const float x2 = px * px + py * py + pz * pz;  // re-added post-reduction
  v2f A;
  A.x = hi ? pz   : px;   // K=2 : K=0
  A.y = hi ? 1.0f : py;   // K=3 : K=1

  float bval[8];
  int   bidx[8];
#pragma unroll
  for (int v = 0; v < 8; ++v) { bval[v] = 3.4e38f; bidx[v] = 0; }

  const v8f Czero = {};   // lowers to inline SRC2 = 0

  auto processTile = [&](float qx, float qy, float qz, int nn) {
    const float y2 = qx * qx + qy * qy + qz * qz;
    v2f Bf;
    Bf.x = hi ? (-2.0f * qz) : (-2.0f * qx);   // K=2 : K=0
    Bf.y = hi ? y2           : (-2.0f * qy);   // K=3 : K=1
    v8f D = __builtin_amdgcn_wmma_f32_16x16x4_f32(
        false, A, false, Bf, (short)0, Czero, false, false);
#pragma unroll
    for (int v = 0; v < 8; ++v) {
      const float d = D[v];
      if (d < bval[v]) { bval[v] = d; bidx[v] = nn; }  // '<' => first index
    }
  };

  // ---- ping-pong double buffer, 2 tiles per buffer, 4 tiles / iteration ----
  const float* ybase = yb + (size_t)(colStart + r) * 3;
  const int last = numTiles - 1;
  // clamped prologue loads (outside the loop; in-bounds for tiny chunks too)
  const int t1c = (1 < last) ? 1 : last;
  const int t2c = (2 < last) ? 2 : last;
  const int t3c = (3 < last) ? 3 : last;
  float a0x = ybase[0],        a0y = ybase[1],         a0z = ybase[2];
  float a1x = ybase[t1c * 48], a1y = ybase[t1c * 48+1], a1z = ybase[t1c * 48+2];
  float b0x = ybase[t2c * 48], b0y = ybase[t2c * 48+1], b0z = ybase[t2c * 48+2];
  float b1x = ybase[t3c * 48], b1y = ybase[t3c * 48+1], b1z = ybase[t3c * 48+2];

  const float* yp = ybase + 4 * 48;   // points at tile t+4
  int n0 = colStart + r;

  // Steady state: branch-free body. Loop runs while >= 8 tiles remain
  // unprocessed; the last 4 tiles always sit in the buffers at exit.
  int t = 0;
  for (; t + 4 < numTiles; t += 4) {
    processTile(a0x, a0y, a0z, n0);
    processTile(a1x, a1y, a1z, n0 + 16);
    a0x = yp[0];  a0y = yp[1];  a0z = yp[2];    // tiles t+4, t+5 -> a*
    a1x = yp[48]; a1y = yp[49]; a1z = yp[50];
    processTile(b0x, b0y, b0z, n0 + 32);
    processTile(b1x, b1y, b1z, n0 + 48);
    b0x = yp[96];  b0y = yp[97];  b0z = yp[98];  // tiles t+6, t+7 -> b*
    b1x = yp[144]; b1y = yp[145]; b1z = yp[146];
    yp += 192;
    n0 += 64;
  }
  // Epilogue: remaining (<= 4) buffered tiles; uniform guards, executed once.
  processTile(a0x, a0y, a0z, n0);
  if (t + 1 < numTiles) processTile(a1x, a1y, a1z, n0 + 16);
  if (t + 2 < numTiles) processTile(b0x, b0y, b0z, n0 + 32);
  if (t + 3 < numTiles) processTile(b1x, b1y, b1z, n0 + 48);

  // ---- butterfly min+argmin across the 16 lanes sharing each row ----
#pragma unroll
  for (int v = 0; v < 8; ++v) {
    float d = bval[v];
    int   i = bidx[v];
#pragma unroll
    for (int mk = 8; mk >= 1; mk >>= 1) {
      const float od = __shfl_xor(d, mk, 32);
      const int   oi = __shfl_xor(i, mk, 32);
      if (od < d || (od == d && oi < i)) { d = od; i = oi; }  // tie -> lower idx
    }
    bval[v] = d;
    bidx[v] = i;
  }

  // lanes 0-15 hold rows rowBase+v, lanes 16-31 hold rows rowBase+8+v.
  // Re-add the row-constant |x_m|^2: writer lane 0 takes row v's x2 from
  // lane v; writer lane 16 takes row 8+v's x2 from lane 8+v.
  const int half = (lane >> 4) << 3;
  float x2row[8];
#pragma unroll
  for (int v = 0; v < 8; ++v) x2row[v] = __shfl(x2, half + v, 32);

  // Merge chunks: packed (orderKey(dist) << 32 | idx) atomic min — total order,
  // deterministic, tie-breaks to the lower column index like jnp.argmin.
  if (lane == 0 || lane == 16) {
    const int mOff = rowBase + half;
#pragma unroll
    for (int v = 0; v < 8; ++v) {
      const float dist = bval[v] + x2row[v];
      const unsigned long long p =
          ((unsigned long long)orderKey(dist) << 32) | (unsigned int)bidx[v];
      atomicMin(&best[(size_t)b * N + mOff + v], p);
    }
  }
}

// ---------------------------------------------------------------------------
__global__ __launch_bounds__(256) void chamfer_init(
    unsigned long long* __restrict__ best, int nTotal, float* __restrict__ out)
{
  const int i = blockIdx.x * blockDim.x + threadIdx.x;
  if (i < nTotal) best[i] = 0xFFFFFFFFFFFFFFFFull;
  if (i < 2) out[i] = 0.0f;
}

// Unpack argmins, gather nearest-neighbor normals, cosine terms, global means.
__global__ __launch_bounds__(256) void chamfer_finalize(
    const unsigned long long* __restrict__ best_x,  // [B,N]
    const unsigned long long* __restrict__ best_y,  // [B,M]
    const float* __restrict__ n1,                   // [B,N,3]
    const float* __restrict__ n2,                   // [B,M,3]
    float* __restrict__ out, int B, int N, int M)
{
  const int total = B * (N + M);
  const int i = blockIdx.x * blockDim.x + threadIdx.x;

  float sd = 0.0f, sn = 0.0f;
  if (i < B * N) {
    const int b = i / N;
    const float invBN = 1.0f / (float)(B * N);
    const unsigned long long p = best_x[i];
    sd = keyToFloat((unsigned int)(p >> 32)) * invBN;
    const int j = (int)(unsigned int)(p & 0xFFFFFFFFu);
    const float* a = n1 + (size_t)i * 3;
    const float* c = n2 + ((size_t)b * M + j) * 3;
    const float dot = a[0]*c[0] + a[1]*c[1] + a[2]*c[2];
    float na = sqrtf(a[0]*a[0] + a[1]*a[1] + a[2]*a[2]);
    float nc = sqrtf(c[0]*c[0] + c[1]*c[1] + c[2]*c[2]);
    na = fmaxf(na, EPSF); nc = fmaxf(nc, EPSF);
    sn = (1.0f - fabsf(dot / (na * nc))) * invBN;
  } else if (i < total) {
    const int k = i - B * N;
    const int b = k / M;
    const float invBM = 1.0f / (float)(B * M);
    const unsigned long long p = best_y[k];
    sd = keyToFloat((unsigned int)(p >> 32)) * invBM;
    const int j = (int)(unsigned int)(p & 0xFFFFFFFFu);
    const float* a = n2 + (size_t)k * 3;
    const float* c = n1 + ((size_t)b * N + j) * 3;
    const float dot = a[0]*c[0] + a[1]*c[1] + a[2]*c[2];
    float na = sqrtf(a[0]*a[0] + a[1]*a[1] + a[2]*a[2]);
    float nc = sqrtf(c[0]*c[0] + c[1]*c[1] + c[2]*c[2]);
    na = fmaxf(na, EPSF); nc = fmaxf(nc, EPSF);
    sn = (1.0f - fabsf(dot / (na * nc))) * invBM;
  }

  __shared__ float s0[256];
  __shared__ float s1[256];
  s0[threadIdx.x] = sd;
  s1[threadIdx.x] = sn;
  __syncthreads();
  for (int off = 128; off > 0; off >>= 1) {
    if (threadIdx.x < off) {
      s0[threadIdx.x] += s0[threadIdx.x + off];
      s1[threadIdx.x] += s1[threadIdx.x + off];
    }
    __syncthreads();
  }
  if (threadIdx.x == 0) {
    atomicAdd(&out[0], s0[0]);
    atomicAdd(&out[1], s1[0]);
  }
}

// ---------------------------------------------------------------------------
extern "C" void kernel_launch(void* const* d_in, const int* in_sizes, int n_in,
                              void* d_out, int out_size, void* d_ws, size_t ws_size,
                              hipStream_t stream) {
  const float* xyz1  = (const float*)d_in[0];
  const float* xyz2  = (const float*)d_in[1];
  const float* nxyz1 = (const float*)d_in[2];
  const float* nxyz2 = (const float*)d_in[3];

  const int B = 4;
  const int N = in_sizes[0] / (B * 3);
  const int M = in_sizes[1] / (B * 3);

  // workspace: packed (key|idx) arrays
  unsigned long long* best_x = (unsigned long long*)d_ws;
  unsigned long long* best_y = best_x + (size_t)B * N;
  float* out = (float*)d_out;

  const int nTotal = B * (N + M);
  chamfer_init<<<(nTotal + 255) / 256, 256, 0, stream>>>(best_x, nTotal, out);

  const int wavesPerBlk = 8;  // 256 threads
  const int wavesX = B * (N / 16) * COLSPLIT;
  const int wavesY = B * (M / 16) * COLSPLIT;

  chamfer_nn_wmma<<<(wavesX + wavesPerBlk - 1) / wavesPerBlk, 256, 0, stream>>>(
      xyz1, xyz2, best_x, N, M, wavesX);
  chamfer_nn_wmma<<<(wavesY + wavesPerBlk - 1) / wavesPerBlk, 256, 0, stream>>>(
      xyz2, xyz1, best_y, M, N, wavesY);

  chamfer_finalize<<<(nTotal + 255) / 256, 256, 0, stream>>>(
      best_x, best_y, nxyz1, nxyz2, out, B, N, M);
}